// MultiChannelClipperModel_54365696033603
// MI455X (gfx1250) — compile-verified
//
#include <hip/hip_runtime.h>
#include <hip/hip_bf16.h>

// Elementwise clamp(x, -0.5f, +0.5f) over 67,108,864 float32 elements.
// HBM-bandwidth-bound (AI = 0.25 FLOP/byte; ~536 MB traffic -> ~23 us floor
// at 23.3 TB/s). Streaming kernel: B128 non-temporal loads/stores (data is
// single-touch; 536 MB footprint > 192 MB L2, so NT avoids cache pollution).
//
// Main kernel is branch-free: each block owns exactly 1024 float4s, each
// thread 4 float4s. All 4 NT B128 loads issue back-to-back (64 B in flight
// per lane) before the first wait, then 4 med3 clamps per vector, then 4 NT
// B128 stores. Tail kernel (guarded) covers any non-multiple remainder.

typedef float v4f __attribute__((ext_vector_type(4)));

#define CLIP_LO (-0.5f)
#define CLIP_HI (0.5f)

#define VEC_PER_THREAD 4
#define BLOCK_THREADS  256
#define VEC_PER_BLOCK  (BLOCK_THREADS * VEC_PER_THREAD)  // 1024 float4s

__device__ __forceinline__ v4f clamp4(v4f v) {
  v.x = fminf(fmaxf(v.x, CLIP_LO), CLIP_HI);
  v.y = fminf(fmaxf(v.y, CLIP_LO), CLIP_HI);
  v.z = fminf(fmaxf(v.z, CLIP_LO), CLIP_HI);
  v.w = fminf(fmaxf(v.w, CLIP_LO), CLIP_HI);
  return v;
}

// Full-tile kernel: no bounds checks, maximal load clustering.
__global__ __launch_bounds__(BLOCK_THREADS) void clip_clamp_main_kernel(
    const v4f* __restrict__ in4, v4f* __restrict__ out4) {
  const unsigned long long base =
      (unsigned long long)blockIdx.x * VEC_PER_BLOCK + threadIdx.x;

  // Issue all loads first so they form one VMEM clause (4x B128 in flight).
  v4f a = __builtin_nontemporal_load(&in4[base + 0ull * BLOCK_THREADS]);
  v4f b = __builtin_nontemporal_load(&in4[base + 1ull * BLOCK_THREADS]);
  v4f c = __builtin_nontemporal_load(&in4[base + 2ull * BLOCK_THREADS]);
  v4f d = __builtin_nontemporal_load(&in4[base + 3ull * BLOCK_THREADS]);

  a = clamp4(a);
  b = clamp4(b);
  c = clamp4(c);
  d = clamp4(d);

  __builtin_nontemporal_store(a, &out4[base + 0ull * BLOCK_THREADS]);
  __builtin_nontemporal_store(b, &out4[base + 1ull * BLOCK_THREADS]);
  __builtin_nontemporal_store(c, &out4[base + 2ull * BLOCK_THREADS]);
  __builtin_nontemporal_store(d, &out4[base + 3ull * BLOCK_THREADS]);
}

// Guarded tail: handles [start_elem, n) scalar elements (covers both the
// leftover float4s and the n%4 remainder). Not launched for this shape.
__global__ void clip_clamp_tail_kernel(const float* __restrict__ in,
                                       float* __restrict__ out,
                                       unsigned long long start,
                                       unsigned long long n) {
  unsigned long long i =
      start + threadIdx.x + (unsigned long long)blockIdx.x * blockDim.x;
  if (i < n) {
    float v = __builtin_nontemporal_load(&in[i]);
    __builtin_nontemporal_store(fminf(fmaxf(v, CLIP_LO), CLIP_HI), &out[i]);
  }
}

extern "C" void kernel_launch(void* const* d_in, const int* in_sizes, int n_in,
                              void* d_out, int out_size, void* d_ws, size_t ws_size,
                              hipStream_t stream) {
  (void)n_in; (void)d_ws; (void)ws_size; (void)out_size;

  const float* x = (const float*)d_in[0];
  float* out = (float*)d_out;
  const unsigned long long n = (unsigned long long)in_sizes[0];

  const unsigned long long n4 = n / 4ull;                    // 16,777,216
  const unsigned long long full_blocks = n4 / VEC_PER_BLOCK; // 16,384
  const unsigned long long covered = full_blocks * (unsigned long long)VEC_PER_BLOCK * 4ull;

  if (full_blocks > 0) {
    clip_clamp_main_kernel<<<(unsigned int)full_blocks, BLOCK_THREADS, 0,
                             stream>>>((const v4f*)x, (v4f*)out);
  }

  if (covered < n) {  // leftover float4s + scalar remainder (0 for this shape)
    const unsigned long long rem = n - covered;
    const unsigned int tblocks = (unsigned int)((rem + 255ull) / 256ull);
    clip_clamp_tail_kernel<<<tblocks, 256, 0, stream>>>(x, out, covered, n);
  }
}